// DeepRTE_81673098100943
// MI455X (gfx1250) — compile-verified
//
#include <hip/hip_runtime.h>
#include <hip/hip_bf16.h>
#include <math.h>

typedef __attribute__((ext_vector_type(2))) float v2f;
typedef __attribute__((ext_vector_type(8))) float v8f;

#define kB   2
#define kP   64
#define kNB  64
#define kT   256
#define kV   32
#define kATT 32
#define kLAT 64

// LDS row strides (floats) — padded to dodge bank conflicts
#define XS 44   // X buffer: 48 rows x 40 cols used
#define KS 36   // Kmat: 48 rows x 32 cols used
#define HS 68   // H buffers: 48 rows x 64 cols used

// ---------------------------------------------------------------------------
// Attention kernel: one wave32 per query row.
// Rows per batch: P (main) + P*V (self) = 2112.  Total waves = 4224.
// ---------------------------------------------------------------------------
__global__ void __launch_bounds__(128) att_kernel(
    const float* __restrict__ phase,   // (B,P,4)
    const float* __restrict__ posc,    // (B,T,2)
    const float* __restrict__ sigma,   // (B,T,2)
    const float* __restrict__ velc,    // (B,V,2)
    const float* __restrict__ Wq, const float* __restrict__ bq,
    const float* __restrict__ Wk, const float* __restrict__ bk,
    const float* __restrict__ Wv, const float* __restrict__ bv,
    const float* __restrict__ Wo, const float* __restrict__ bo,
    float* __restrict__ attMain,       // (B,P,32)
    float* __restrict__ attSelf)       // (B,P,V,32)
{
  const int lane = threadIdx.x & 31;
  const int wid  = (blockIdx.x * blockDim.x + threadIdx.x) >> 5;
  const int rowsPerB = kP + kP * kV;
  const int b = wid / rowsPerB;
  const int m = wid % rowsPerB;

  float c0, c1, c2, c3;
  float* dst;
  if (m < kP) {
    const float* cc = phase + (b * kP + m) * 4;
    c0 = cc[0]; c1 = cc[1]; c2 = cc[2]; c3 = cc[3];
    dst = attMain + (b * kP + m) * kATT;
  } else {
    int idx = m - kP;
    int p = idx / kV;
    int v = idx % kV;
    const float* cc = phase + (b * kP + p) * 4;
    c0 = cc[0]; c1 = cc[1];
    c2 = velc[(b * kV + v) * 2 + 0];
    c3 = velc[(b * kV + v) * 2 + 1];
    dst = attSelf + ((b * kP + p) * kV + v) * kATT;
  }

  // q = coord @ Wq + bq  (wave-uniform, 32 values)
  float q[32];
  #pragma unroll
  for (int j = 0; j < 32; j++)
    q[j] = bq[j] + c0 * Wq[j] + c1 * Wq[32 + j] + c2 * Wq[64 + j] + c3 * Wq[96 + j];

  const float inv_sqrt_kd = 0.35355339059327373f;  // 1/sqrt(8)
  float lg[4][8];

  #pragma unroll
  for (int tt = 0; tt < 8; tt++) {
    int t = tt * 32 + lane;
    float px = posc[(b * kT + t) * 2 + 0];
    float py = posc[(b * kT + t) * 2 + 1];
    float r0 = px - c0, r1 = py - c1;           // rel = position - pos
    bool keep = (r0 * c2 + r1 * c3) <= 0.0f;    // mask: rel . vel <= 0
    float l0 = 0.f, l1 = 0.f, l2 = 0.f, l3 = 0.f;
    #pragma unroll
    for (int j = 0; j < 32; j++) {
      // k_j = local @ Wk + bk, local = [rel0, rel1, vel0, vel1]
      float kj = bk[j] + r0 * Wk[j] + r1 * Wk[32 + j] + c2 * Wk[64 + j] + c3 * Wk[96 + j];
      float qk = q[j] * kj;
      if (j < 8) l0 += qk; else if (j < 16) l1 += qk; else if (j < 24) l2 += qk; else l3 += qk;
    }
    lg[0][tt] = keep ? l0 * inv_sqrt_kd : -1e30f;
    lg[1][tt] = keep ? l1 * inv_sqrt_kd : -1e30f;
    lg[2][tt] = keep ? l2 * inv_sqrt_kd : -1e30f;
    lg[3][tt] = keep ? l3 * inv_sqrt_kd : -1e30f;
  }

  // masked softmax over T per head (wave reduction across 32 lanes x 8 slots)
  float mx[4], sm[4];
  #pragma unroll
  for (int h = 0; h < 4; h++) {
    float m0 = -3.4e38f;
    #pragma unroll
    for (int tt = 0; tt < 8; tt++) m0 = fmaxf(m0, lg[h][tt]);
    #pragma unroll
    for (int off = 16; off > 0; off >>= 1) m0 = fmaxf(m0, __shfl_xor(m0, off, 32));
    mx[h] = m0;
    float s0 = 0.f;
    #pragma unroll
    for (int tt = 0; tt < 8; tt++) s0 += expf(lg[h][tt] - m0);
    #pragma unroll
    for (int off = 16; off > 0; off >>= 1) s0 += __shfl_xor(s0, off, 32);
    sm[h] = s0;
  }

  // attn = softmax(w) @ v, v = sigma @ Wv + bv
  float acc[32];
  #pragma unroll
  for (int j = 0; j < 32; j++) acc[j] = 0.f;
  #pragma unroll
  for (int tt = 0; tt < 8; tt++) {
    int t = tt * 32 + lane;
    float s0v = sigma[(b * kT + t) * 2 + 0];
    float s1v = sigma[(b * kT + t) * 2 + 1];
    float w0 = expf(lg[0][tt] - mx[0]) / sm[0];
    float w1 = expf(lg[1][tt] - mx[1]) / sm[1];
    float w2 = expf(lg[2][tt] - mx[2]) / sm[2];
    float w3 = expf(lg[3][tt] - mx[3]) / sm[3];
    #pragma unroll
    for (int j = 0; j < 32; j++) {
      float vj = bv[j] + s0v * Wv[j] + s1v * Wv[32 + j];
      float w = (j < 8) ? w0 : (j < 16) ? w1 : (j < 24) ? w2 : w3;
      acc[j] += w * vj;
    }
  }
  #pragma unroll
  for (int j = 0; j < 32; j++) {
    #pragma unroll
    for (int off = 16; off > 0; off >>= 1) acc[j] += __shfl_xor(acc[j], off, 32);
  }

  // out_j = exp(-(attn @ Wo + bo)_j); lane j computes element j
  float o = bo[lane];
  #pragma unroll
  for (int i = 0; i < 32; i++) o += acc[i] * Wo[i * 32 + lane];
  dst[lane] = expf(-o);
}

// ---------------------------------------------------------------------------
// WMMA GEMM helper: 48x64 output (3 waves, each wave = 16x64 tile, 4 v8f accs)
// D = A(48xK) * B(Kx64) + bias, optional tanh, optional residual add.
// A lane layout (f32 16x4): lanes 0-15 -> K={k0,k0+1}, lanes 16-31 -> K={k0+2,k0+3}
// B lane layout (f32 4x16): symmetric (row K striped across lanes within VGPR)
// C/D layout: vgpr r, lanes 0-15: M=r, N=lane ; lanes 16-31: M=r+8, N=lane-16
// ---------------------------------------------------------------------------
__device__ __forceinline__ void gemm48(
    const float* sIn, int inStride, int K,
    const float* Bm, int bStride,
    const float* bias,
    float* sOut, int outStride,
    bool doTanh, bool residual)
{
  const int lane = threadIdx.x & 31;
  const int wave = threadIdx.x >> 5;
  const int col  = lane & 15;
  const int half = lane >> 4;           // 0 or 1
  const int aRow = wave * 16 + col;     // A-frag M index
  const int koff = half * 2;

  v8f acc[4];
  #pragma unroll
  for (int c = 0; c < 4; c++) {
    float bvv = bias ? bias[c * 16 + col] : 0.0f;
    #pragma unroll
    for (int r = 0; r < 8; r++) acc[c][r] = bvv;
  }

  for (int k0 = 0; k0 < K; k0 += 4) {
    v2f a;
    a.x = sIn[aRow * inStride + k0 + koff];
    a.y = sIn[aRow * inStride + k0 + koff + 1];
    #pragma unroll
    for (int c = 0; c < 4; c++) {
      v2f bf;
      bf.x = Bm[(k0 + koff) * bStride + c * 16 + col];
      bf.y = Bm[(k0 + koff + 1) * bStride + c * 16 + col];
      acc[c] = __builtin_amdgcn_wmma_f32_16x16x4_f32(
          false, a, false, bf, (short)0, acc[c], false, false);
    }
  }

  const int mBase = wave * 16 + half * 8;
  #pragma unroll
  for (int c = 0; c < 4; c++) {
    #pragma unroll
    for (int r = 0; r < 8; r++) {
      float vv = acc[c][r];
      if (doTanh) vv = tanhf(vv);
      float* pdst = &sOut[(mBase + r) * outStride + c * 16 + col];
      if (residual) *pdst += vv; else *pdst = vv;
    }
  }
}

// ---------------------------------------------------------------------------
// Per-(b,p,n) block: 33 MLP rows (1 act + 32 self_act) padded to 48,
// 3-layer tanh MLP + 2 mixing rounds + green output (pre-weighted by wq).
// ---------------------------------------------------------------------------
__global__ void __launch_bounds__(96) block_kernel(
    const float* __restrict__ phase,    // (B,P,4)
    const float* __restrict__ bcoord,   // (B,NB,4)
    const float* __restrict__ boundary, // (B,NB)
    const float* __restrict__ bweights, // (B,NB)
    const float* __restrict__ velc,     // (B,V,2)
    const float* __restrict__ velw,     // (B,V)
    const float* __restrict__ scatk,    // (B,P,V)
    const float* __restrict__ selfk,    // (B,V,V)
    const float* __restrict__ W1, const float* __restrict__ b1,
    const float* __restrict__ W2, const float* __restrict__ b2,
    const float* __restrict__ Wout, const float* __restrict__ bout,
    const float* __restrict__ Ws1, const float* __restrict__ bs1,
    const float* __restrict__ Ws2, const float* __restrict__ bs2,
    const float* __restrict__ Wp,       // (64,1)
    const float* __restrict__ attMain,  // (B,P,32)
    const float* __restrict__ attSelf,  // (B,P,V,32)
    float* __restrict__ greenOut)       // (B,P,NB), includes wq factor
{
  __shared__ float sX[48 * XS];
  __shared__ float sK[48 * KS];
  __shared__ float sH0[48 * HS];
  __shared__ float sH1[48 * HS];

  const int tid = threadIdx.x;
  const int lane = tid & 31;
  const int wave = tid >> 5;

  const int b  = blockIdx.x / (kP * kNB);
  const int pn = blockIdx.x % (kP * kNB);
  const int p  = pn / kNB;
  const int n  = pn % kNB;

  // ---- stage 0: build X (48x40) and Kmat (48x32) in LDS ----
  for (int i = tid; i < 48 * XS; i += 96) sX[i] = 0.f;
  for (int i = tid; i < 48 * KS; i += 96) sK[i] = 0.f;
  __syncthreads();

  for (int i = tid; i < 33 * 40; i += 96) {
    int row = i / 40, c = i % 40;
    float val;
    if (row == 0) {
      if (c < 4)       val = phase[(b * kP + p) * 4 + c];
      else if (c < 8)  val = bcoord[(b * kNB + n) * 4 + (c - 4)];
      else             val = attMain[(b * kP + p) * kATT + (c - 8)];
    } else {
      int v = row - 1;
      if (c < 2)       val = phase[(b * kP + p) * 4 + c];
      else if (c < 4)  val = velc[(b * kV + v) * 2 + (c - 2)];
      else if (c < 8)  val = bcoord[(b * kNB + n) * 4 + (c - 4)];
      else             val = attSelf[((b * kP + p) * kV + v) * kATT + (c - 8)];
    }
    sX[row * XS + c] = val;
  }
  for (int i = tid; i < 33 * 32; i += 96) {
    int row = i / 32, j = i % 32;
    float vw = velw[b * kV + j];
    float val = (row == 0)
        ? (1.0f - scatk[(b * kP + p) * kV + j]) * vw
        : (1.0f - selfk[(b * kV + (row - 1)) * kV + j]) * vw;
    sK[row * KS + j] = val;
  }
  __syncthreads();

  // ---- MLP: tanh(X@W1+b1) -> tanh(@W2+b2) -> tanh(@Wout+bout) ----
  gemm48(sX,  XS, 40, W1,   64, b1,   sH0, HS, true, false);
  __syncthreads();
  gemm48(sH0, HS, 64, W2,   64, b2,   sH1, HS, true, false);
  __syncthreads();
  gemm48(sH1, HS, 64, Wout, 64, bout, sH0, HS, true, false);
  __syncthreads();
  // sH0 now holds: row 0 = act, rows 1..32 = self_act (rows 33..47 junk)

  // ---- two mixing rounds ----
  #pragma unroll
  for (int round = 0; round < 2; round++) {
    const float* Ws = (round == 0) ? Ws1 : Ws2;
    const float* bs = (round == 0) ? bs1 : bs2;
    // mix = Kmat(48x32) @ self_act(32x64)   (self_act = sH0 rows 1..32)
    gemm48(sK, KS, 32, sH0 + HS, HS, nullptr, sH1, HS, false, false);
    __syncthreads();
    // act += tanh(mix @ Ws + bs)
    gemm48(sH1, HS, 64, Ws, 64, bs, sH0, HS, true, true);
    __syncthreads();
  }

  // ---- green = exp(act_row0 . Wp); scale by wq and store ----
  if (wave == 0) {
    float part = sH0[lane] * Wp[lane] + sH0[32 + lane] * Wp[32 + lane];
    #pragma unroll
    for (int off = 16; off > 0; off >>= 1) part += __shfl_xor(part, off, 32);
    if (lane == 0) {
      float wq = boundary[b * kNB + n] * bweights[b * kNB + n];
      greenOut[blockIdx.x] = expf(part) * wq;
    }
  }
}

// ---------------------------------------------------------------------------
// Deterministic reduction over NB per (b,p)
// ---------------------------------------------------------------------------
__global__ void __launch_bounds__(128) finalize_kernel(
    const float* __restrict__ green, float* __restrict__ out)
{
  int i = threadIdx.x;   // 0..127 = b*P + p
  float s = 0.f;
  #pragma unroll 4
  for (int nn = 0; nn < kNB; nn++) s += green[i * kNB + nn];
  out[i] = s;
}

// ---------------------------------------------------------------------------
extern "C" void kernel_launch(void* const* d_in, const int* in_sizes, int n_in,
                              void* d_out, int out_size, void* d_ws, size_t ws_size,
                              hipStream_t stream) {
  const float* phase    = (const float*)d_in[0];
  const float* bcoord   = (const float*)d_in[1];
  const float* boundary = (const float*)d_in[2];
  const float* bweights = (const float*)d_in[3];
  const float* posc     = (const float*)d_in[4];
  const float* sigma    = (const float*)d_in[5];
  const float* velc     = (const float*)d_in[6];
  const float* velw     = (const float*)d_in[7];
  const float* scatk    = (const float*)d_in[8];
  const float* selfk    = (const float*)d_in[9];
  const float* Wq   = (const float*)d_in[10];
  const float* bq   = (const float*)d_in[11];
  const float* Wk   = (const float*)d_in[12];
  const float* bk   = (const float*)d_in[13];
  const float* Wv   = (const float*)d_in[14];
  const float* bv   = (const float*)d_in[15];
  const float* Wo   = (const float*)d_in[16];
  const float* bo   = (const float*)d_in[17];
  const float* W1   = (const float*)d_in[18];
  const float* b1   = (const float*)d_in[19];
  const float* W2   = (const float*)d_in[20];
  const float* b2   = (const float*)d_in[21];
  const float* Wout = (const float*)d_in[22];
  const float* bout = (const float*)d_in[23];
  const float* Ws1  = (const float*)d_in[24];
  const float* bs1  = (const float*)d_in[25];
  const float* Ws2  = (const float*)d_in[26];
  const float* bs2  = (const float*)d_in[27];
  const float* Wp   = (const float*)d_in[28];

  float* ws = (float*)d_ws;
  float* attMain = ws;                                  // B*P*32      = 4096
  float* attSelf = attMain + kB * kP * kATT;            // B*P*V*32    = 131072
  float* green   = attSelf + kB * kP * kV * kATT;       // B*P*NB      = 8192

  // 4224 waves total / 4 waves per block
  att_kernel<<<1056, 128, 0, stream>>>(
      phase, posc, sigma, velc, Wq, bq, Wk, bk, Wv, bv, Wo, bo,
      attMain, attSelf);

  block_kernel<<<kB * kP * kNB, 96, 0, stream>>>(
      phase, bcoord, boundary, bweights, velc, velw, scatk, selfk,
      W1, b1, W2, b2, Wout, bout, Ws1, bs1, Ws2, bs2, Wp,
      attMain, attSelf, green);

  finalize_kernel<<<1, 128, 0, stream>>>(green, (float*)d_out);
}